// PillarFeatureNet_69741678953059
// MI455X (gfx1250) — compile-verified
//
#include <hip/hip_runtime.h>
#include <hip/hip_bf16.h>

typedef __attribute__((ext_vector_type(2))) float v2f;
typedef __attribute__((ext_vector_type(8))) float v8f;

#define NSTAT 27            // 6 first moments + 21 unique second moments
#define STATS_OFF 0
#define WP_OFF    32        // 64x6 folded weights
#define BP_OFF    (32 + 64*6)

__device__ __forceinline__ void atomAddF32(float* p, float v) {
    unsafeAtomicAdd(p, v);  // native global_atomic_add_f32 path on gfx1250
}

// ---------------------------------------------------------------- zero kernel
__global__ void pfn_zero_kernel(float4* __restrict__ out4, int n4,
                                float* __restrict__ out, int n,
                                float* __restrict__ stats) {
    int gtid = blockIdx.x * blockDim.x + threadIdx.x;
    if (blockIdx.x == 0 && threadIdx.x < 32) stats[threadIdx.x] = 0.0f;
    int stride = gridDim.x * blockDim.x;
    for (int i = gtid; i < n4; i += stride) out4[i] = make_float4(0.f, 0.f, 0.f, 0.f);
    int r = n4 * 4 + gtid;
    if (r < n) out[r] = 0.0f;  // tail (out_size not /4)
}

// ------------------------------------------------------- x statistics kernel
// Accumulates sum(x_k) and sum(x_k * x_l) (k<=l) over all points.
__global__ void pfn_stats_kernel(const float* __restrict__ x, int N,
                                 float* __restrict__ stats) {
    __shared__ float red[256 * NSTAT];
    float s[NSTAT];
#pragma unroll
    for (int j = 0; j < NSTAT; ++j) s[j] = 0.0f;

    int stride = gridDim.x * blockDim.x;
    for (int p = blockIdx.x * blockDim.x + threadIdx.x; p < N; p += stride) {
        const float* xr = x + (size_t)p * 6;
        float v[6];
#pragma unroll
        for (int k = 0; k < 6; ++k) v[k] = xr[k];
#pragma unroll
        for (int k = 0; k < 6; ++k) s[k] += v[k];
        int t = 6;
#pragma unroll
        for (int k = 0; k < 6; ++k)
#pragma unroll
            for (int l = k; l < 6; ++l) s[t++] += v[k] * v[l];
    }
#pragma unroll
    for (int j = 0; j < NSTAT; ++j) red[threadIdx.x * NSTAT + j] = s[j];
    __syncthreads();
    if (threadIdx.x < NSTAT) {
        float a = 0.0f;
        for (int i = 0; i < 256; ++i) a += red[i * NSTAT + threadIdx.x];
        atomAddF32(&stats[threadIdx.x], a);
    }
}

// --------------------------------------------- fold BN into weights (64 thr)
__global__ void pfn_finalize_kernel(const float* __restrict__ stats,
                                    const float* __restrict__ W,
                                    const float* __restrict__ b,
                                    const float* __restrict__ gamma,
                                    const float* __restrict__ beta,
                                    float* __restrict__ Wp,
                                    float* __restrict__ bp, float invN) {
    int j = threadIdx.x;  // 0..63
    float Ex[6], Exx[6][6];
#pragma unroll
    for (int k = 0; k < 6; ++k) Ex[k] = stats[k] * invN;
    int t = 6;
#pragma unroll
    for (int k = 0; k < 6; ++k)
#pragma unroll
        for (int l = k; l < 6; ++l) {
            float v = stats[t++] * invN;
            Exx[k][l] = v; Exx[l][k] = v;
        }
    float wj[6];
#pragma unroll
    for (int k = 0; k < 6; ++k) wj[k] = W[j * 6 + k];
    float bj = b[j];

    float wEx = 0.0f;
#pragma unroll
    for (int k = 0; k < 6; ++k) wEx += wj[k] * Ex[k];
    float mean = bj + wEx;

    float q = 0.0f;
#pragma unroll
    for (int k = 0; k < 6; ++k)
#pragma unroll
        for (int l = 0; l < 6; ++l) q += wj[k] * wj[l] * Exx[k][l];
    float Eh2 = q + 2.0f * bj * wEx + bj * bj;
    float var = Eh2 - mean * mean;
    var = fmaxf(var, 0.0f);
    float sc = gamma[j] * rsqrtf(var + 1e-5f);
#pragma unroll
    for (int k = 0; k < 6; ++k) Wp[j * 6 + k] = wj[k] * sc;
    bp[j] = bj * sc + beta[j] - mean * sc;
}

// ---------------------------------------------- fused WMMA + ReLU + scatter
// Per wave: 16-point tile. A = x tile (16x4 f32, 2 k-steps for K=6 padded to 8),
// B = W'^T chunk (4x16 f32), C/D = 16x16 f32.  4 N-chunks x 2 k-steps = 8 WMMAs.
__global__ void pfn_main_kernel(const float* __restrict__ x,
                                const int* __restrict__ indices,
                                const float* __restrict__ Wp,
                                const float* __restrict__ bp,
                                float* __restrict__ grid, int N) {
    const int lane  = threadIdx.x & 31;
    const int wave  = threadIdx.x >> 5;
    const int wpb   = blockDim.x >> 5;
    const int gw    = blockIdx.x * wpb + wave;
    const int nwave = gridDim.x * wpb;

    const int n     = lane & 15;        // feature within chunk / point row
    const int khalf = lane >> 4;        // 0: K=0,1(k-step0) | 1: K=2,3

    // ---- loop-invariant B (W'^T) and bias, resident in VGPRs -------------
    v2f  B0[4], B1[4];
    float bias[4];
#pragma unroll
    for (int c = 0; c < 4; ++c) {
        int col = c * 16 + n;                 // output feature
        // k-step 0: rows K = {0,1} (lanes 0-15) / {2,3} (lanes 16-31)
        B0[c].x = Wp[col * 6 + 2 * khalf];
        B0[c].y = Wp[col * 6 + 2 * khalf + 1];
        // k-step 1: rows K = {4,5} valid, {6,7} are zero padding
        if (khalf == 0) { B1[c].x = Wp[col * 6 + 4]; B1[c].y = Wp[col * 6 + 5]; }
        else            { B1[c].x = 0.0f;            B1[c].y = 0.0f;            }
        bias[c] = bp[col];
    }

    const int nTiles = N >> 4;
    for (int tile = gw; tile < nTiles; tile += nwave) {
        // ---- A matrix: 16 points x K (padded) ----------------------------
        const float* xr = x + (size_t)(tile * 16 + n) * 6;
        v2f a0, a1;
        a0.x = xr[2 * khalf];
        a0.y = xr[2 * khalf + 1];
        if (khalf == 0) { a1.x = xr[4]; a1.y = xr[5]; }
        else            { a1.x = 0.0f;  a1.y = 0.0f;  }

        // flat pillar index for "my" point (m = lane&15)
        int p  = tile * 16 + n;
        int fx = indices[2 * p];
        int fy = indices[2 * p + 1];
        int flat = fx * 512 + fy;

#pragma unroll
        for (int c = 0; c < 4; ++c) {
            v8f acc;
#pragma unroll
            for (int r = 0; r < 8; ++r) acc[r] = bias[c];
            acc = __builtin_amdgcn_wmma_f32_16x16x4_f32(
                false, a0, false, B0[c], (short)0, acc, false, false);
            acc = __builtin_amdgcn_wmma_f32_16x16x4_f32(
                false, a1, false, B1[c], (short)0, acc, false, false);
            // ReLU + scatter.  VGPR r holds point M = r + 8*khalf, feature
            // N = c*16 + n  (C/D layout per ISA 7.12.2).
#pragma unroll
            for (int r = 0; r < 8; ++r) {
                float v  = fmaxf(acc[r], 0.0f);
                int   m2 = r + (khalf << 3);
                int   fm = __shfl(flat, m2, 32);
                atomAddF32(grid + (size_t)fm * 64 + c * 16 + n, v);
            }
        }
    }

    // ---- tail points (N not multiple of 16): plain VALU path -------------
    if (gw == 0) {
        for (int p = nTiles * 16 + lane; p < N; p += 32) {
            const float* xr = x + (size_t)p * 6;
            float v[6];
#pragma unroll
            for (int k = 0; k < 6; ++k) v[k] = xr[k];
            int flat = indices[2 * p] * 512 + indices[2 * p + 1];
            for (int j = 0; j < 64; ++j) {
                float a = bp[j];
#pragma unroll
                for (int k = 0; k < 6; ++k) a += Wp[j * 6 + k] * v[k];
                a = fmaxf(a, 0.0f);
                atomAddF32(grid + (size_t)flat * 64 + j, a);
            }
        }
    }
}

// ----------------------------------------------------------------- launcher
extern "C" void kernel_launch(void* const* d_in, const int* in_sizes, int n_in,
                              void* d_out, int out_size, void* d_ws, size_t ws_size,
                              hipStream_t stream) {
    const float* x       = (const float*)d_in[0];
    const int*   indices = (const int*)d_in[1];
    const float* W       = (const float*)d_in[2];
    const float* b       = (const float*)d_in[3];
    const float* gamma   = (const float*)d_in[4];
    const float* beta    = (const float*)d_in[5];

    int N = in_sizes[0] / 6;

    float* grid  = (float*)d_out;
    float* ws    = (float*)d_ws;
    float* stats = ws + STATS_OFF;
    float* Wp    = ws + WP_OFF;
    float* bp    = ws + BP_OFF;

    int n4 = out_size >> 2;
    pfn_zero_kernel<<<2048, 256, 0, stream>>>((float4*)d_out, n4, grid, out_size, stats);
    pfn_stats_kernel<<<256, 256, 0, stream>>>(x, N, stats);
    pfn_finalize_kernel<<<1, 64, 0, stream>>>(stats, W, b, gamma, beta, Wp, bp,
                                              1.0f / (float)N);
    pfn_main_kernel<<<1024, 256, 0, stream>>>(x, indices, Wp, bp, grid, N);
}